// Gene2SubDifferCrossAttn_42073499631971
// MI455X (gfx1250) — compile-verified
//
#include <hip/hip_runtime.h>
#include <math.h>

typedef float v2f __attribute__((ext_vector_type(2)));
typedef float v8f __attribute__((ext_vector_type(8)));

#define WMMA_F32(a, b, c) \
  __builtin_amdgcn_wmma_f32_16x16x4_f32(false, (a), false, (b), (short)0, (c), false, false)

static constexpr int B_ = 8, S_ = 512, E_ = 32, L_ = 4096, SUBE_ = 768;
static constexpr float LAMBDA_INIT = 0.2f;   // 0.8 - 0.6*exp(0)

// Head-0 (diff) LDS row stride padded to 513 so the WMMA A-fragment reads
// (lane n -> row n) hit 16 distinct banks.  Head-1 region stays at 512
// (only touched with stride-1-across-lanes patterns).
static constexpr int ST0 = 513;              // head0 / diff row stride (floats)
static constexpr int H1  = 16 * ST0;         // head1 region offset (floats)
static constexpr int SMEM_FLOATS = 16 * ST0 + 16 * 512;   // 65,600 bytes

// ---------------------------------------------------------------------------
// Kernel 1: K/V projection.  Km[bs, o] = sum_e key[bs, e] * Wk[o, e]  (same Wv)
// GEMM M=B*S=4096, N=32 (2 tiles), K=768 (192 k-steps of 4). One wave / M-tile.
// ---------------------------------------------------------------------------
__global__ __launch_bounds__(32) void kv_proj_kernel(
    const float* __restrict__ key, const float* __restrict__ Wk,
    const float* __restrict__ Wv, float* __restrict__ Km, float* __restrict__ Vm) {
  const int lane = threadIdx.x;
  const int n    = lane & 15;
  const int khi  = lane >> 4;
  const int m0   = blockIdx.x * 16;          // flat row tile over B*S
  const int mrow = m0 + n;                   // A-frag row for this lane

  v8f ak0 = {}, ak1 = {}, av0 = {}, av1 = {};
  for (int kk = 0; kk < SUBE_ / 4; ++kk) {
    const int kb = kk * 4 + khi * 2;
    v2f a   = *(const v2f*)(key + (size_t)mrow * SUBE_ + kb);
    v2f bk0 = *(const v2f*)(Wk + (size_t)n * SUBE_ + kb);
    v2f bk1 = *(const v2f*)(Wk + (size_t)(16 + n) * SUBE_ + kb);
    v2f bv0 = *(const v2f*)(Wv + (size_t)n * SUBE_ + kb);
    v2f bv1 = *(const v2f*)(Wv + (size_t)(16 + n) * SUBE_ + kb);
    ak0 = WMMA_F32(a, bk0, ak0);
    ak1 = WMMA_F32(a, bk1, ak1);
    av0 = WMMA_F32(a, bv0, av0);
    av1 = WMMA_F32(a, bv1, av1);
  }
  const int rbase = m0 + (khi ? 8 : 0);
  for (int v = 0; v < 8; ++v) {
    const size_t r = (size_t)(rbase + v) * 32;
    Km[r + n]      = ak0[v];
    Km[r + 16 + n] = ak1[v];
    Vm[r + n]      = av0[v];
    Vm[r + 16 + n] = av1[v];
  }
}

// ---------------------------------------------------------------------------
// Kernel 2: Q projection, pre-scaled by H^-0.5 = 0.25.
// GEMM M=B*L=32768, N=32, K=32 (8 k-steps). One wave / M-tile.
// ---------------------------------------------------------------------------
__global__ __launch_bounds__(32) void q_proj_kernel(
    const float* __restrict__ q, const float* __restrict__ Wq,
    float* __restrict__ Qs) {
  const int lane = threadIdx.x;
  const int n    = lane & 15;
  const int khi  = lane >> 4;
  const int m0   = blockIdx.x * 16;
  const int mrow = m0 + n;

  v8f a0 = {}, a1 = {};
  for (int kk = 0; kk < 8; ++kk) {
    const int kb = kk * 4 + khi * 2;
    v2f a  = *(const v2f*)(q + (size_t)mrow * 32 + kb);
    v2f b0 = *(const v2f*)(Wq + (size_t)n * 32 + kb);
    v2f b1 = *(const v2f*)(Wq + (size_t)(16 + n) * 32 + kb);
    a0 = WMMA_F32(a, b0, a0);
    a1 = WMMA_F32(a, b1, a1);
  }
  const int rbase = m0 + (khi ? 8 : 0);
  for (int v = 0; v < 8; ++v) {
    const size_t r = (size_t)(rbase + v) * 32;
    Qs[r + n]      = a0[v] * 0.25f;
    Qs[r + 16 + n] = a1[v] * 0.25f;
  }
}

// ---------------------------------------------------------------------------
// Kernel 3: fused differential attention for one 16-row L-tile.
// LDS: head0 scores/diff [16][513] + head1 scores [16][512]  (65,600 B).
// Head1 region is reused for the 16x32 attn_out tile.
// ---------------------------------------------------------------------------
__global__ __launch_bounds__(32) void attn_kernel(
    const float* __restrict__ Qs, const float* __restrict__ Km,
    const float* __restrict__ Vm, const int* __restrict__ qmask,
    const int* __restrict__ kmask, const float* __restrict__ lq1,
    const float* __restrict__ lk1, const float* __restrict__ lq2,
    const float* __restrict__ lk2, const float* __restrict__ lnw,
    const float* __restrict__ Wo, float* __restrict__ outp,
    float* __restrict__ diffp) {
  __shared__ float smem[SMEM_FLOATS];

  const int lane = threadIdx.x;
  const int n    = lane & 15;
  const int khi  = lane >> 4;
  const int R0   = blockIdx.x * 16;          // flat row base over B*L
  const int b    = R0 / L_;
  const int rb   = khi ? 8 : 0;              // C/D row base for this lane half

  // lambda_full = exp(lq1.lk1) - exp(lq2.lk2) + lambda_init (redundant per lane)
  float s1 = 0.f, s2 = 0.f;
  for (int i = 0; i < 16; ++i) { s1 += lq1[i] * lk1[i]; s2 += lq2[i] * lk2[i]; }
  const float lambda_full = __expf(s1) - __expf(s2) + LAMBDA_INIT;

  // ---- scores = Q*K^T per head, masked, into LDS -------------------------
  for (int h = 0; h < 2; ++h) {
    float* dst       = (h == 0) ? smem : (smem + H1);
    const int stride = (h == 0) ? ST0 : 512;
    for (int st = 0; st < 32; ++st) {
      v8f acc = {};
      for (int kk = 0; kk < 4; ++kk) {
        const int kb = kk * 4 + khi * 2;
        v2f a  = *(const v2f*)(Qs + (size_t)(R0 + n) * 32 + h * 16 + kb);
        v2f bf = *(const v2f*)(Km + (size_t)(b * S_ + st * 16 + n) * 32 + h * 16 + kb);
        acc = WMMA_F32(a, bf, acc);
      }
      const int col = st * 16 + n;
      const int kmv = kmask[b * S_ + col];
      for (int v = 0; v < 8; ++v) {
        const int r     = rb + v;
        const int valid = kmv && qmask[R0 + r];
        dst[r * stride + col] = valid ? acc[v] : -1e20f;
      }
    }
  }
  __syncthreads();

  // ---- per-row softmax(2 heads) + differential combine + min-shift + mask
  for (int m = 0; m < 16; ++m) {
    const float* r0 = smem + m * ST0;
    const float* r1 = smem + H1 + m * 512;
    float mx0 = -3.4e38f, mx1 = -3.4e38f;
    for (int i = 0; i < 16; ++i) {
      const int c = lane + 32 * i;
      mx0 = fmaxf(mx0, r0[c]);
      mx1 = fmaxf(mx1, r1[c]);
    }
    for (int o = 16; o > 0; o >>= 1) {
      mx0 = fmaxf(mx0, __shfl_xor(mx0, o, 32));
      mx1 = fmaxf(mx1, __shfl_xor(mx1, o, 32));
    }
    float e0[16], e1[16], sum0 = 0.f, sum1 = 0.f;
    for (int i = 0; i < 16; ++i) {
      const int c = lane + 32 * i;
      e0[i] = __expf(r0[c] - mx0); sum0 += e0[i];
      e1[i] = __expf(r1[c] - mx1); sum1 += e1[i];
    }
    for (int o = 16; o > 0; o >>= 1) {
      sum0 += __shfl_xor(sum0, o, 32);
      sum1 += __shfl_xor(sum1, o, 32);
    }
    const float inv0 = 1.f / (sum0 + 1e-8f);
    const float inv1 = 1.f / (sum1 + 1e-8f);
    float d[16], dmin = 3.4e38f;
    for (int i = 0; i < 16; ++i) {
      d[i] = e0[i] * inv0 - lambda_full * (e1[i] * inv1);
      dmin = fminf(dmin, d[i]);
    }
    for (int o = 16; o > 0; o >>= 1) dmin = fminf(dmin, __shfl_xor(dmin, o, 32));

    const int qv = qmask[R0 + m];
    float* w0 = smem + m * ST0;                       // overwrite with diff
    float* dg = diffp + (size_t)(R0 + m) * 512;       // coalesced 128B stores
    for (int i = 0; i < 16; ++i) {
      const int c = lane + 32 * i;
      float dd = d[i] - dmin + 1e-5f;
      dd = (qv && kmask[b * S_ + c]) ? dd : 0.f;
      w0[c] = dd;
      __builtin_nontemporal_store(dd, dg + c);        // 64MB stream: bypass L2 RT
    }
  }
  __syncthreads();

  // ---- attn_out = diff(16x512) x V(512x32) via WMMA (128 k-steps) --------
  // A-frag LDS reads: lane n -> smem[n*513 + kb]: 16 distinct banks.
  v8f ao0 = {}, ao1 = {};
  for (int kk = 0; kk < 128; ++kk) {
    const int kb = kk * 4 + khi * 2;
    v2f a; a[0] = smem[n * ST0 + kb]; a[1] = smem[n * ST0 + kb + 1];
    const float* vr0 = Vm + (size_t)(b * S_ + kb) * 32;
    const float* vr1 = Vm + (size_t)(b * S_ + kb + 1) * 32;
    v2f b0; b0[0] = vr0[n];      b0[1] = vr1[n];
    v2f b1; b1[0] = vr0[16 + n]; b1[1] = vr1[16 + n];
    ao0 = WMMA_F32(a, b0, ao0);
    ao1 = WMMA_F32(a, b1, ao1);
  }
  float* aos = smem + H1;                    // reuse head1 region: [16][32]
  for (int v = 0; v < 8; ++v) {
    const int r = rb + v;
    aos[r * 32 + n]      = ao0[v];
    aos[r * 32 + 16 + n] = ao1[v];
  }
  __syncthreads();

  // ---- RMSNorm over E=32 + affine + (1 - lambda_init) --------------------
  for (int m = 0; m < 16; ++m) {
    const float x = aos[m * 32 + lane];
    float sq = x * x;
    for (int o = 16; o > 0; o >>= 1) sq += __shfl_xor(sq, o, 32);
    const float rms = rsqrtf(sq * (1.0f / 32.0f) + 1e-5f);
    aos[m * 32 + lane] = x * rms * lnw[lane] * 0.8f;
  }
  __syncthreads();

  // ---- output projection: (16x32) x Wo^T(32x32) via WMMA -----------------
  v8f o0 = {}, o1 = {};
  for (int kk = 0; kk < 8; ++kk) {
    const int kb = kk * 4 + khi * 2;
    v2f a; a[0] = aos[n * 32 + kb]; a[1] = aos[n * 32 + kb + 1];
    v2f b0 = *(const v2f*)(Wo + (size_t)n * 32 + kb);
    v2f b1 = *(const v2f*)(Wo + (size_t)(16 + n) * 32 + kb);
    o0 = WMMA_F32(a, b0, o0);
    o1 = WMMA_F32(a, b1, o1);
  }
  for (int v = 0; v < 8; ++v) {
    const size_t r = (size_t)(R0 + rb + v) * 32;
    __builtin_nontemporal_store(o0[v], outp + r + n);
    __builtin_nontemporal_store(o1[v], outp + r + 16 + n);
  }
}

// ---------------------------------------------------------------------------
extern "C" void kernel_launch(void* const* d_in, const int* in_sizes, int n_in,
                              void* d_out, int out_size, void* d_ws, size_t ws_size,
                              hipStream_t stream) {
  (void)in_sizes; (void)n_in; (void)out_size; (void)ws_size;
  const float* query = (const float*)d_in[0];   // [B,P,G,32] -> [B,L,32]
  const float* key   = (const float*)d_in[1];   // [B,S,768]
  const int*   qmask = (const int*)d_in[2];     // [B,P,G] -> [B,L]
  const int*   kmask = (const int*)d_in[3];     // [B,S]
  const float* Wq = (const float*)d_in[4];
  const float* Wk = (const float*)d_in[5];
  const float* Wv = (const float*)d_in[6];
  const float* Wo = (const float*)d_in[7];
  const float* lq1 = (const float*)d_in[8];
  const float* lk1 = (const float*)d_in[9];
  const float* lq2 = (const float*)d_in[10];
  const float* lk2 = (const float*)d_in[11];
  const float* lnw = (const float*)d_in[12];

  float* outp  = (float*)d_out;                        // B*L*32
  float* diffp = outp + (size_t)B_ * L_ * E_;          // B*L*S

  float* Qs = (float*)d_ws;                            // B*L*32  (4 MB)
  float* Km = Qs + (size_t)B_ * L_ * E_;               // B*S*32  (0.5 MB)
  float* Vm = Km + (size_t)B_ * S_ * E_;               // B*S*32  (0.5 MB)

  kv_proj_kernel<<<(B_ * S_) / 16, 32, 0, stream>>>(key, Wk, Wv, Km, Vm);
  q_proj_kernel<<<(B_ * L_) / 16, 32, 0, stream>>>(query, Wq, Qs);
  attn_kernel<<<(B_ * L_) / 16, 32, 0, stream>>>(Qs, Km, Vm, qmask, kmask,
                                                 lq1, lk1, lq2, lk2, lnw, Wo,
                                                 outp, diffp);
}